// GemmaAttention_47236050321952
// MI455X (gfx1250) — compile-verified
//
#include <hip/hip_runtime.h>
#include <hip/hip_bf16.h>
#include <math.h>

typedef __attribute__((ext_vector_type(2))) float v2f;
typedef __attribute__((ext_vector_type(4))) float v4f;
typedef __attribute__((ext_vector_type(8))) float v8f;

#define B_     2
#define S_     2048
#define H_     16
#define KVH_   8
#define D_     128
#define HID_   2048
#define NREP_  2
#define SCALING_ 0.08838834764831845f   // 128^-0.5
#define SOFTCAP_ 50.0f
#define NEGINF_ (-3.4028235e38f)

static __device__ __forceinline__ v8f wmma_f32(v2f a, v2f b, v8f c) {
  // D = A(16x4) * B(4x16) + C, full fp32
  return __builtin_amdgcn_wmma_f32_16x16x4_f32(false, a, false, b, (short)0, c, false, false);
}

static __device__ __forceinline__ float rmax16(float v) {
  for (int off = 8; off; off >>= 1) v = fmaxf(v, __shfl_xor(v, off, 16));
  return v;
}
static __device__ __forceinline__ float rsum16(float v) {
  for (int off = 8; off; off >>= 1) v += __shfl_xor(v, off, 16);
  return v;
}

// Low 32 bits of a generic pointer to __shared__ = LDS byte offset (ISA 10.2).
static __device__ __forceinline__ unsigned lds_off(const void* p) {
  return (unsigned)(size_t)p;
}
// Async global->LDS copy, 16B per lane. GVS addressing: saddr + voffset.
static __device__ __forceinline__ void async_copy_b128(unsigned ldsoff,
                                                       const void* sbase,
                                                       unsigned voff) {
  asm volatile("global_load_async_to_lds_b128 %0, %1, %2"
               :: "v"(ldsoff), "v"(voff), "s"(sbase) : "memory");
}
static __device__ __forceinline__ void wait_async0() {
  asm volatile("s_wait_asynccnt 0" ::: "memory");
}

// C[M,N] = A[M,K] @ W[N,K]^T + bias[N]   (fp32 WMMA 16x16x4, async double-buffered LDS)
__global__ __launch_bounds__(256)
void gemm_xwT_kernel(const float* __restrict__ A, const float* __restrict__ W,
                     const float* __restrict__ bias, float* __restrict__ C,
                     int M, int N, int K) {
  __shared__ float As[2][128][36];   // [buf][m][k]  (pad 36)
  __shared__ float Bs[2][128][36];   // [buf][n][k]  (row-major, NO transpose)

  const int t = threadIdx.x;
  const int lane = t & 31, wave = t >> 5;
  const int half = lane >> 4, nn = lane & 15, kb = half * 2;
  const int wr = wave >> 2, wc = wave & 3;            // 2 x 4 wave grid
  const int m0 = blockIdx.y * 128, n0 = blockIdx.x * 128;

  // per-thread staging coordinates (4 x 16B each for A and for W)
  const int srow = t >> 3;            // 0..31 step over 128 rows via +32*i? no: id-based below
  (void)srow;

  v8f zero = {};
  v8f acc[4][2];
  for (int i = 0; i < 4; ++i)
    for (int j = 0; j < 2; ++j) acc[i][j] = zero;

  auto stage = [&](int buf, int k0) {
    const float* abase = A + k0;
    const float* wbase = W + k0;
    for (int i = 0; i < 4; ++i) {
      int id = t + i * 256;                 // 1024 float4 per tile
      int row = id >> 3, c4 = (id & 7) << 2;
      async_copy_b128(lds_off(&As[buf][row][c4]), abase,
                      (unsigned)(((size_t)(m0 + row) * K + c4) * 4));
      async_copy_b128(lds_off(&Bs[buf][row][c4]), wbase,
                      (unsigned)(((size_t)(n0 + row) * K + c4) * 4));
    }
  };

  stage(0, 0);
  for (int k0 = 0; k0 < K; k0 += 32) {
    const int cur = (k0 >> 5) & 1;
    wait_async0();          // this wave's tile (and prior) landed in LDS
    __syncthreads();        // everyone's tile landed; prior buffer free to reuse
    if (k0 + 32 < K) stage(cur ^ 1, k0 + 32);

    // software-pipelined 8 x (K=4) WMMA steps
    v2f a[2][4], b[2][2];
    auto ldfrags = [&](int kk, v2f* av, v2f* bv) {
      for (int rt = 0; rt < 4; ++rt)
        av[rt] = *reinterpret_cast<const v2f*>(&As[cur][wr * 64 + rt * 16 + nn][kk + kb]);
      for (int ct = 0; ct < 2; ++ct)
        bv[ct] = *reinterpret_cast<const v2f*>(&Bs[cur][wc * 32 + ct * 16 + nn][kk + kb]);
    };
    ldfrags(0, a[0], b[0]);
#pragma unroll
    for (int s = 0; s < 8; ++s) {
      if (s < 7) ldfrags((s + 1) * 4, a[(s + 1) & 1], b[(s + 1) & 1]);
      for (int rt = 0; rt < 4; ++rt)
        for (int ct = 0; ct < 2; ++ct)
          acc[rt][ct] = wmma_f32(a[s & 1][rt], b[s & 1][ct], acc[rt][ct]);
    }
  }

  for (int rt = 0; rt < 4; ++rt)
    for (int ct = 0; ct < 2; ++ct)
      for (int r = 0; r < 8; ++r) {
        int m = m0 + wr * 64 + rt * 16 + r + half * 8;   // C layout: row = r + 8*half
        int n = n0 + wc * 32 + ct * 16 + nn;
        C[(size_t)m * N + n] = acc[rt][ct][r] + bias[n];
      }
}

// RoPE on q/k, fold D^-0.5 into q, scatter to [B,H,S,D]/[B,KVH,S,D] layouts.
__global__ __launch_bounds__(256)
void rope_split_kernel(const float* __restrict__ qkv, const float* __restrict__ cosb,
                       const float* __restrict__ sinb, float* __restrict__ Qr,
                       float* __restrict__ Kr, float* __restrict__ Vr) {
  const int PAIRS = (H_ + 2 * KVH_) * D_ / 2;   // 2048 pairs per (b,s)
  int idx = blockIdx.x * blockDim.x + threadIdx.x;
  if (idx >= B_ * S_ * PAIRS) return;
  int pair = idx % PAIRS;
  int bs = idx / PAIRS;
  int b = bs / S_, s = bs % S_;
  int c = pair * 2;
  const float* row = qkv + (size_t)bs * ((H_ + 2 * KVH_) * D_);

  if (c < H_ * D_) {
    int h = c >> 7, j = (c & 127) >> 1;
    float xr = row[c], xi = row[c + 1];
    float cv = cosb[s * (D_ / 2) + j], sv = sinb[s * (D_ / 2) + j];
    float* dst = Qr + ((size_t)(b * H_ + h) * S_ + s) * D_ + 2 * j;
    dst[0] = (xr * cv - xi * sv) * SCALING_;
    dst[1] = (xr * sv + xi * cv) * SCALING_;
  } else if (c < (H_ + KVH_) * D_) {
    int cc = c - H_ * D_;
    int h = cc >> 7, j = (cc & 127) >> 1;
    float xr = row[c], xi = row[c + 1];
    float cv = cosb[s * (D_ / 2) + j], sv = sinb[s * (D_ / 2) + j];
    float* dst = Kr + ((size_t)(b * KVH_ + h) * S_ + s) * D_ + 2 * j;
    dst[0] = xr * cv - xi * sv;
    dst[1] = xr * sv + xi * cv;
  } else {
    int cc = c - (H_ + KVH_) * D_;
    int h = cc >> 7, d = cc & 127;
    float* dst = Vr + ((size_t)(b * KVH_ + h) * S_ + s) * D_ + d;
    dst[0] = row[c];
    dst[1] = row[c + 1];
  }
}

// Flash attention: grid (B*H, S/64), 4 waves, 16 q-rows per wave.
__global__ __launch_bounds__(128)
void flash_attn_kernel(const float* __restrict__ Qr, const float* __restrict__ Kr,
                       const float* __restrict__ Vr, float* __restrict__ Ao) {
  __shared__ float Qs[64][132];      // [q][d]
  __shared__ float Ks[16][132];      // [k][d]
  __shared__ float VsT[128][20];     // [d][k]  (transposed for contiguous B-frags)
  __shared__ float Ps[4][16][20];    // per-wave P tile [m][k]

  const int t = threadIdx.x;
  const int lane = t & 31, wave = t >> 5;
  const int half = lane >> 4, nn = lane & 15, kb = half * 2;
  const int bh = blockIdx.x;
  const int b = bh / H_, h = bh % H_;
  const int kvh = h / NREP_;
  const int q0 = blockIdx.y * 64;

  const float* Qbase = Qr + (size_t)(b * H_ + h) * S_ * D_;
  const float* Kbase = Kr + (size_t)(b * KVH_ + kvh) * S_ * D_;
  const float* Vbase = Vr + (size_t)(b * KVH_ + kvh) * S_ * D_;

  // async-stage Q tile (64 x 128) once
  {
    const float* qb = Qbase + (size_t)q0 * D_;
    for (int i = 0; i < 16; ++i) {
      int id = t + i * 128;                 // 2048 float4
      int row = id >> 5, c4 = (id & 31) << 2;
      async_copy_b128(lds_off(&Qs[row][c4]), qb,
                      (unsigned)(((size_t)row * D_ + c4) * 4));
    }
  }

  v8f zero = {};
  v8f acc[8];
  for (int i = 0; i < 8; ++i) acc[i] = zero;
  float m_run[8], l_run[8];
  for (int r = 0; r < 8; ++r) { m_run[r] = NEGINF_; l_run[r] = 0.0f; }

  const int nkt = q0 / 16 + 4;            // causal: k <= q0+63
  for (int kt = 0; kt < nkt; ++kt) {
    int k0 = kt * 16;
    __syncthreads();                      // prior iter done with Ks/VsT
    // K tile: async contiguous copy
    {
      const float* kbp = Kbase + (size_t)k0 * D_;
      for (int i = 0; i < 4; ++i) {
        int id = t + i * 128;             // 512 float4
        int row = id >> 5, c4 = (id & 31) << 2;
        async_copy_b128(lds_off(&Ks[row][c4]), kbp,
                        (unsigned)(((size_t)row * D_ + c4) * 4));
      }
    }
    // V tile: register-mediated transpose into VsT[d][k]
    for (int i = 0; i < 4; ++i) {
      int id = t + i * 128;
      int row = id >> 5, c4 = (id & 31) << 2;
      v4f v = *reinterpret_cast<const v4f*>(&Vbase[(size_t)(k0 + row) * D_ + c4]);
      VsT[c4 + 0][row] = v.x;
      VsT[c4 + 1][row] = v.y;
      VsT[c4 + 2][row] = v.z;
      VsT[c4 + 3][row] = v.w;
    }
    wait_async0();                        // own Q/K asyncs landed
    __syncthreads();

    // S(16x16) = Qwave(16x128) @ K(16x128)^T : 32 fp32 WMMAs over D
    v8f s = zero;
#pragma unroll
    for (int d0 = 0; d0 < D_; d0 += 4) {
      v2f a = *reinterpret_cast<const v2f*>(&Qs[wave * 16 + nn][d0 + kb]);
      v2f bf = *reinterpret_cast<const v2f*>(&Ks[nn][d0 + kb]);
      s = wmma_f32(a, bf, s);
    }

    // softcap + causal mask + online softmax (row = r + 8*half, col = nn)
    for (int r = 0; r < 8; ++r) {
      int qrow = q0 + wave * 16 + r + half * 8;
      int kcol = k0 + nn;
      float sv = SOFTCAP_ * tanhf(s[r] * (1.0f / SOFTCAP_));
      sv = (kcol > qrow) ? NEGINF_ : sv;
      float mt = rmax16(sv);
      float mn = fmaxf(m_run[r], mt);
      float alpha = __expf(m_run[r] - mn);
      float p = __expf(sv - mn);
      float lt = rsum16(p);
      l_run[r] = l_run[r] * alpha + lt;
      m_run[r] = mn;
      for (int ct = 0; ct < 8; ++ct) acc[ct][r] *= alpha;
      Ps[wave][r + half * 8][nn] = p;     // P in [m][k] order
    }
    asm volatile("s_wait_dscnt 0" ::: "memory");   // same-wave LDS RAW on Ps

    // O(16x128) += P(16x16) @ V(16x128) : 32 fp32 WMMAs, contiguous frags
#pragma unroll
    for (int ct = 0; ct < 8; ++ct) {
#pragma unroll
      for (int kk = 0; kk < 16; kk += 4) {
        v2f a = *reinterpret_cast<const v2f*>(&Ps[wave][nn][kk + kb]);
        v2f bf = *reinterpret_cast<const v2f*>(&VsT[ct * 16 + nn][kk + kb]);
        acc[ct] = wmma_f32(a, bf, acc[ct]);
      }
    }
  }

  // normalize + store to Ao[b][q][h*128 + d]
  for (int r = 0; r < 8; ++r) {
    int qrow = q0 + wave * 16 + r + half * 8;
    float inv = 1.0f / l_run[r];
    for (int ct = 0; ct < 8; ++ct) {
      Ao[((size_t)b * S_ + qrow) * (H_ * D_) + h * D_ + ct * 16 + nn] = acc[ct][r] * inv;
    }
  }
}

extern "C" void kernel_launch(void* const* d_in, const int* in_sizes, int n_in,
                              void* d_out, int out_size, void* d_ws, size_t ws_size,
                              hipStream_t stream) {
  (void)in_sizes; (void)n_in; (void)out_size; (void)ws_size;
  const float* hidden = (const float*)d_in[0];
  const float* cosb   = (const float*)d_in[1];
  const float* sinb   = (const float*)d_in[2];
  const float* qkv_w  = (const float*)d_in[3];
  const float* qkv_b  = (const float*)d_in[4];
  const float* o_w    = (const float*)d_in[5];
  const float* o_b    = (const float*)d_in[6];
  float* out = (float*)d_out;

  char* ws = (char*)d_ws;
  const size_t QKV_BYTES = (size_t)B_ * S_ * (H_ + 2 * KVH_) * D_ * 4;  // 67,108,864
  const size_t Q_BYTES   = (size_t)B_ * H_ * S_ * D_ * 4;               // 33,554,432
  const size_t K_BYTES   = (size_t)B_ * KVH_ * S_ * D_ * 4;             // 16,777,216
  float* qkv = (float*)ws;
  float* Qr  = (float*)(ws + QKV_BYTES);
  float* Kr  = (float*)(ws + QKV_BYTES + Q_BYTES);
  float* Vr  = (float*)(ws + QKV_BYTES + Q_BYTES + K_BYTES);
  float* Ao  = qkv;   // qkv buffer dead after rope; reuse for attention output

  const int M = B_ * S_;                       // 4096
  // 1) QKV projection: [4096,2048] @ [4096,2048]^T
  gemm_xwT_kernel<<<dim3((H_ + 2 * KVH_) * D_ / 128, M / 128), 256, 0, stream>>>(
      hidden, qkv_w, qkv_b, qkv, M, (H_ + 2 * KVH_) * D_, HID_);
  // 2) RoPE + layout split
  int pairs = B_ * S_ * ((H_ + 2 * KVH_) * D_ / 2);
  rope_split_kernel<<<(pairs + 255) / 256, 256, 0, stream>>>(qkv, cosb, sinb, Qr, Kr, Vr);
  // 3) causal flash attention with tanh softcap
  flash_attn_kernel<<<dim3(B_ * H_, S_ / 64), 128, 0, stream>>>(Qr, Kr, Vr, Ao);
  // 4) output projection: [4096,2048] @ [2048,2048]^T
  gemm_xwT_kernel<<<dim3(HID_ / 128, M / 128), 256, 0, stream>>>(
      Ao, o_w, o_b, out, M, HID_, HID_);
}